// custom_loss_22660247454107
// MI455X (gfx1250) — compile-verified
//
#include <hip/hip_runtime.h>
#include <math.h>

#define BATCH 4096
#define VOCAB 50257
#define TPB   256          // 8 waves of 32 on gfx1250

typedef float v2f __attribute__((ext_vector_type(2)));
typedef float v4f __attribute__((ext_vector_type(4)));
typedef float v8f __attribute__((ext_vector_type(8)));

// ---- toolchain probes (answers show up in compiler stderr) ----
#if defined(__AMDGCN__)
#if __has_builtin(__builtin_amdgcn_wmma_f32_16x16x4_f32)
#pragma message("PROBE: wmma_f32_16x16x4_f32 PRESENT")
#else
#pragma message("PROBE: wmma_f32_16x16x4_f32 ABSENT")
#endif
#if __has_builtin(__builtin_amdgcn_global_load_async_to_lds_b128)
#pragma message("PROBE: global_load_async_to_lds_b128 PRESENT")
#else
#pragma message("PROBE: global_load_async_to_lds_b128 ABSENT")
#endif
#if __has_builtin(__builtin_amdgcn_s_wait_asynccnt)
#pragma message("PROBE: s_wait_asynccnt PRESENT")
#else
#pragma message("PROBE: s_wait_asynccnt ABSENT")
#endif
#if __has_builtin(__builtin_amdgcn_cluster_load_async_to_lds_b128)
#pragma message("PROBE: cluster_load_async_to_lds_b128 PRESENT")
#endif
#endif

// online logsumexp: push one element
__device__ __forceinline__ void lse_push(float& m, float& s, float x) {
    float nm = fmaxf(m, x);
    s = s * __expf(m - nm) + __expf(x - nm);
    m = nm;
}

// merge two (m,s) states (both must have seen >=1 element)
__device__ __forceinline__ void lse_combine(float& m, float& s, float mo, float so) {
    float nm = fmaxf(m, mo);
    s = s * __expf(m - nm) + so * __expf(mo - nm);
    m = nm;
}

// One block per row: single-pass online logsumexp over 50257 fp32 logits.
__global__ void __launch_bounds__(TPB) row_lse_kernel(
    const float* __restrict__ logits,
    const int*   __restrict__ target,
    float*       __restrict__ row_loss)
{
    const int row = blockIdx.x;
    const int tid = threadIdx.x;
    const float* rowp = logits + (size_t)row * (size_t)VOCAB;

    // picked logit: only needed by thread 0 (issue the load early)
    float picked = 0.0f;
    if (tid == 0) picked = rowp[target[row]];

    // peel to 16B alignment (row base is only 4B aligned since VOCAB is odd)
    const int head = (int)(((16u - ((unsigned)(uintptr_t)rowp & 15u)) & 15u) >> 2);
    const int nv   = (VOCAB - head) >> 2;          // float4 body count
    const int tail = VOCAB - head - (nv << 2);     // 0..3 scalars at the end
    const v4f* body = (const v4f*)(rowp + head);

    // 4 independent accumulators -> breaks the serial max/exp chain
    float m0 = -INFINITY, m1 = -INFINITY, m2 = -INFINITY, m3 = -INFINITY;
    float s0 = 0.f, s1 = 0.f, s2 = 0.f, s3 = 0.f;

    const int PF = TPB * 6;                        // prefetch ~24KB ahead
    for (int i = tid; i < nv; i += TPB) {
        __builtin_prefetch(body + i + PF, 0, 3);   // global_prefetch_b8
        v4f v = body[i];
        lse_push(m0, s0, v[0]);
        lse_push(m1, s1, v[1]);
        lse_push(m2, s2, v[2]);
        lse_push(m3, s3, v[3]);
    }

    // head + tail scalars handled by thread 0
    if (tid == 0) {
        for (int i = 0; i < head; ++i)  lse_push(m0, s0, rowp[i]);
        for (int i = 0; i < tail; ++i)  lse_push(m0, s0, rowp[head + (nv << 2) + i]);
    }

    // fold the 4 lanes-of-ILP accumulators
    lse_combine(m0, s0, m1, s1);
    lse_combine(m2, s2, m3, s3);
    lse_combine(m0, s0, m2, s2);

    // wave32 reduction
    float m = m0, s = s0;
    #pragma unroll
    for (int off = 16; off > 0; off >>= 1) {
        float mo = __shfl_xor(m, off);
        float so = __shfl_xor(s, off);
        lse_combine(m, s, mo, so);
    }

    __shared__ float sm[TPB / 32];
    __shared__ float ss[TPB / 32];
    if ((tid & 31) == 0) { sm[tid >> 5] = m; ss[tid >> 5] = s; }
    __syncthreads();

    if (tid < 32) {
        const int idx = tid & 7;                   // replicate so all lanes hold real data
        float mf = sm[idx], sf = ss[idx];
        #pragma unroll
        for (int off = 4; off > 0; off >>= 1) {
            float mo = __shfl_xor(mf, off);
            float so = __shfl_xor(sf, off);
            lse_combine(mf, sf, mo, so);
        }
        if (tid == 0) {
            float logz = mf + __logf(sf);
            row_loss[row] = logz - picked;
        }
    }
}

// Deterministic sum of the 4096 per-row losses.
// WMMA path: D = A(16x4 chunk) * B(ones 4x16) + C accumulates row sums into the
// f32 C tile on one wave (EXEC all-1s inside the tid<32 branch); 64 chained WMMAs.
__global__ void __launch_bounds__(TPB) final_reduce_kernel(
    const float* __restrict__ rl,
    float*       __restrict__ out)
{
#if __has_builtin(__builtin_amdgcn_wmma_f32_16x16x4_f32)
    if (threadIdx.x < 32) {
        const int lane = threadIdx.x;
        // A-matrix 16x4 f32 layout: lanes 0-15 hold K=0,1 ; lanes 16-31 hold K=2,3
        const int mrow = lane & 15;
        const int koff = (lane >> 4) * 2;
        const float* base = rl + mrow * 4 + koff;  // 8B aligned (even offset)

        v8f c = {0.f, 0.f, 0.f, 0.f, 0.f, 0.f, 0.f, 0.f};
        v2f ones = {1.0f, 1.0f};                   // B layout irrelevant: all ones

        for (int ch = 0; ch < BATCH / 64; ++ch) {  // 64 chunks of 64 values
            v2f a = *(const v2f*)(base + ch * 64);
            c = __builtin_amdgcn_wmma_f32_16x16x4_f32(
                    false, a, false, ones, (short)0, c, false, false);
        }
        // column N=lane of C holds row sums: M=0..7 in c[0..7] (lanes 0-15),
        // M=8..15 in c[0..7] (lanes 16-31). Take column 0 = lane0 + lane16.
        float p = ((c[0] + c[1]) + (c[2] + c[3])) + ((c[4] + c[5]) + (c[6] + c[7]));
        float tot = __shfl(p, 0) + __shfl(p, 16);
        if (lane == 0) out[0] = tot;
    }
#else
    // deterministic LDS tree fallback (same <<<1,TPB>>> launch shape)
    __shared__ float buf[TPB];
    const int per = BATCH / TPB;                   // 16
    float s = 0.f;
    for (int k = 0; k < per; ++k) s += rl[threadIdx.x * per + k];
    buf[threadIdx.x] = s;
    __syncthreads();
    for (int off = TPB / 2; off > 0; off >>= 1) {
        if (threadIdx.x < off) buf[threadIdx.x] += buf[threadIdx.x + off];
        __syncthreads();
    }
    if (threadIdx.x == 0) out[0] = buf[0];
#endif
}

extern "C" void kernel_launch(void* const* d_in, const int* in_sizes, int n_in,
                              void* d_out, int out_size, void* d_ws, size_t ws_size,
                              hipStream_t stream)
{
    const float* logits = (const float*)d_in[0];
    const int*   target = (const int*)d_in[1];
    float*       row_loss = (float*)d_ws;          // 4096 floats = 16KB scratch

    row_lse_kernel<<<BATCH, TPB, 0, stream>>>(logits, target, row_loss);
    final_reduce_kernel<<<1, TPB, 0, stream>>>(row_loss, (float*)d_out);
}